// LinearCRF_74775380623454
// MI455X (gfx1250) — compile-verified
//
#include <hip/hip_runtime.h>
#include <stdint.h>

// Problem dims (fixed by the reference)
#define B_  512
#define S_  1024
#define L_  64
#define NT  256                       // 4 threads per tag column

// Feature streaming: CH steps per async chunk, double buffered
#define CH   8
#define NC   (S_ / CH)                // 128 chunks

// LDS layout (dynamic shared; ~70KB/workgroup, needs CDNA5's 320KB WGP LDS)
#define BP_OFF     0u                 // 1023*64 bytes of backpointers
#define FV_OFF     65536u             // 2*64 floats ping-pong forward_var
#define FEAT_OFF   66048u             // 2 chunks of CH*64 floats
#define CHBYTES    (CH * L_ * 4u)     // 2048 bytes per chunk
#define SMEM_BYTES (FEAT_OFF + 2u * CHBYTES)   // 70144 bytes

// One async chunk (CH*64 floats = 2KB): one global_load_async_to_lds_b64 per
// thread (256 threads x 8B). One ASYNCcnt increment per wave per chunk.
__device__ __forceinline__ void async_issue_chunk(const float* gbase,
                                                  uint32_t lds_base,
                                                  int tid) {
  uint32_t l = lds_base + (uint32_t)tid * 8u;
  uint64_t g = (uint64_t)(uintptr_t)(gbase + tid * 2);
  asm volatile("global_load_async_to_lds_b64 %0, %1, off"
               :: "v"(l), "v"(g) : "memory");
}

// (max, first-index) merge: strictly greater wins; equal -> smaller index.
__device__ __forceinline__ void amerge(float om, int oa, float& m, int& a) {
  bool take = (om > m) || ((om == m) && (oa < a));
  a = take ? oa : a;
  m = take ? om : m;
}

__global__ void __launch_bounds__(NT)
viterbi_crf_kernel(const float* __restrict__ features,     // [B,S,L] f32
                   const float* __restrict__ transitions,  // [L,L]   f32
                   float* __restrict__ out)                // [B,S]
{
  extern __shared__ unsigned char smem[];
  unsigned char* bp = smem + BP_OFF;                 // [1023][64] uint8
  float* fv = (float*)(smem + FV_OFF);               // [2][64] ping-pong

  const int tid = threadIdx.x;
  const int j   = tid >> 2;          // tag column
  const int q   = tid & 3;           // quarter of the i-range: [16q, 16q+16)
  const int b   = blockIdx.x;
  const float* featb = features + (size_t)b * S_ * L_;

  // My 16 transition values T[16q+i, j] pinned in VGPRs.
  float tcol[16];
#pragma unroll
  for (int i = 0; i < 16; ++i)
    tcol[i] = transitions[(q * 16 + i) * L_ + j];

  // Prime the async feature pipeline: chunks 0 and 1.
  async_issue_chunk(featb,           FEAT_OFF,           tid);
  async_issue_chunk(featb + CH * L_, FEAT_OFF + CHBYTES, tid);

  int p = 0;  // fv[p] holds current forward_var
#pragma unroll 1
  for (int c = 0; c < NC; ++c) {
    // Wait for chunk c (1 async instr per wave per chunk may stay in flight).
    if (c + 1 < NC) { asm volatile("s_wait_asynccnt 1" ::: "memory"); }
    else            { asm volatile("s_wait_asynccnt 0" ::: "memory"); }
    __syncthreads();   // publish async LDS writes from all waves

    const float* fb = (const float*)(smem + FEAT_OFF + (uint32_t)(c & 1) * CHBYTES);

#pragma unroll 1
    for (int tt = 0; tt < CH; ++tt) {
      const int t = c * CH + tt;
      if (t == 0) {
        if (q == 0) fv[j] = fb[tt * L_ + j];         // forward_var = f[0]
      } else {
        const float* fvp  = fv + p * L_ + q * 16;
        const int ibase   = q * 16;
        // 4 independent sub-chains over my 16 elements (short dep chains)
        float m0, m1, m2, m3;
        int   a0, a1, a2, a3;
        {
          const float4 v = *(const float4*)(fvp);
          m0 = v.x + tcol[0]; a0 = ibase + 0;
          m1 = v.y + tcol[1]; a1 = ibase + 1;
          m2 = v.z + tcol[2]; a2 = ibase + 2;
          m3 = v.w + tcol[3]; a3 = ibase + 3;
        }
#pragma unroll
        for (int k = 1; k < 4; ++k) {
          const float4 v = *(const float4*)(fvp + 4 * k);
          float s; bool g;
          s = v.x + tcol[4*k+0]; g = s > m0; a0 = g ? ibase+4*k+0 : a0; m0 = g ? s : m0;
          s = v.y + tcol[4*k+1]; g = s > m1; a1 = g ? ibase+4*k+1 : a1; m1 = g ? s : m1;
          s = v.z + tcol[4*k+2]; g = s > m2; a2 = g ? ibase+4*k+2 : a2; m2 = g ? s : m2;
          s = v.w + tcol[4*k+3]; g = s > m3; a3 = g ? ibase+4*k+3 : a3; m3 = g ? s : m3;
        }
        float m = m0; int a = a0;
        amerge(m1, a1, m, a);
        amerge(m2, a2, m, a);
        amerge(m3, a3, m, a);
        // Butterfly across the 4 wave-adjacent lanes of this column.
#pragma unroll
        for (int d = 1; d <= 2; d <<= 1) {
          float om = __shfl_xor(m, d, 32);
          int   oa = __shfl_xor(a, d, 32);
          amerge(om, oa, m, a);
        }
        if (q == 0) {
          bp[(t - 1) * L_ + j] = (unsigned char)a;   // backpointer in LDS
          fv[(p ^ 1) * L_ + j] = fb[tt * L_ + j] + m;
        }
        p ^= 1;
      }
      __syncthreads();
    }

    // All threads are past chunk c's reads -> safe to overwrite its buffer.
    if (c + 2 < NC)
      async_issue_chunk(featb + (size_t)(c + 2) * CH * L_,
                        FEAT_OFF + (uint32_t)((c + 2) & 1) * CHBYTES, tid);
  }

  // Backtrack entirely in LDS (first-max tie-break like jnp.argmax).
  if (tid == 0) {
    const float* fvp = fv + p * L_;
    float best = fvp[0];
    int tag = 0;
    for (int jj = 1; jj < L_; ++jj) {
      float v = fvp[jj];
      if (v > best) { best = v; tag = jj; }
    }
    float* ob = out + (size_t)b * S_;
    ob[S_ - 1] = (float)tag;
    for (int t = S_ - 2; t >= 0; --t) {
      tag = bp[t * L_ + tag];
      ob[t] = (float)tag;
    }
  }
}

extern "C" void kernel_launch(void* const* d_in, const int* in_sizes, int n_in,
                              void* d_out, int out_size, void* d_ws, size_t ws_size,
                              hipStream_t stream) {
  (void)in_sizes; (void)n_in; (void)out_size; (void)d_ws; (void)ws_size;
  const float* features    = (const float*)d_in[0];   // [512,1024,64] f32
  const float* transitions = (const float*)d_in[1];   // [64,64] f32
  float* out = (float*)d_out;                         // [512,1024]
  viterbi_crf_kernel<<<dim3(B_), dim3(NT), SMEM_BYTES, stream>>>(
      features, transitions, out);
}